// CenterNetDecode_80049600463592
// MI455X (gfx1250) — compile-verified
//
#include <hip/hip_runtime.h>
#include <stdint.h>

// CenterNet decode for MI455X (gfx1250).
// Bandwidth-bound streaming kernel:
//  - heat read once via async global->LDS (ASYNCcnt path), 4 channels per
//    LDS buffer, double-buffered -> 8 outstanding loads/wave, 40 barriers.
//  - box tile fetched by the Tensor Data Mover (TENSORcnt path), overlapping
//    the whole 80-channel loop.

namespace {
constexpr int kB  = 8;
constexpr int kC  = 80;
constexpr int kH  = 192;
constexpr int kW  = 192;
constexpr int kTX = 32;             // tile width  (pixels)
constexpr int kTY = 8;              // tile height (pixels)
constexpr int kHaloW = kTX + 2;     // 34
constexpr int kHaloH = kTY + 2;     // 10
constexpr int kHaloN = kHaloW * kHaloH;   // 340 halo cells
constexpr int kSlots = 512;         // padded: every thread owns exactly 2 slots
constexpr uint32_t kBufBytes = kSlots * 4;     // 2048 B per channel slice
constexpr int kCPB = 4;                        // channels per LDS buffer
constexpr int kGroups = kC / kCPB;             // 20
constexpr uint32_t kGrpBytes = kCPB * kBufBytes; // 8192 B per group buffer
constexpr float kR   = 4.0f;        // DOWNSAMPLE
constexpr float kThr = 0.7f;        // THRESHOLD
} // namespace

typedef __attribute__((ext_vector_type(4))) unsigned int u32x4;
typedef __attribute__((ext_vector_type(8))) int          i32x8;
typedef __attribute__((ext_vector_type(4))) int          i32x4;

// Zero 256 B of workspace each launch (deterministic, capture-safe) so OOB
// halo cells have a valid zero page to async-load from.
__global__ void cn_zero_ws(float* ws) {
  if (threadIdx.x < 64) ws[threadIdx.x] = 0.0f;
}

// Raw LDS byte address (addrspace(3) ptrtoint) as required by the async/TDM
// LDS-address operands.
__device__ __forceinline__ uint32_t lds_addr32(void* p) {
  return (uint32_t)(uintptr_t)(__attribute__((address_space(3))) char*)p;
}

__global__ __launch_bounds__(256, 4) void cn_decode(
    const float* __restrict__ heat,   // [8,80,192,192]
    const float* __restrict__ box,    // [8,4,192,192]
    float* __restrict__ out,          // boxes | mask | score | center (flat)
    const float* __restrict__ zeropage) {
  __shared__ float shHeat[2][kCPB][kSlots];   // 16 KB, double-buffered groups
  __shared__ float shBox[4][kTY][kTX];        // 4 KB, filled by TDM

  const int tid    = threadIdx.x;
  const int tileX0 = blockIdx.x * kTX;
  const int tileY0 = blockIdx.y * kTY;
  const int b      = blockIdx.z;

  const size_t plane = (size_t)kH * kW; // 36864

  // ---------------- TDM: DMA the box tile (32x8x4 planes) to LDS -----------
  // D# per CDNA5 ISA sec.8: count=1, type=2 ("image"),
  // data_size=4B, tensor dims 192x192(x4), tile 32x8x4, stride0=192,
  // stride1=36864 (plane). Tile is always fully in-bounds. Issued once by
  // wave 0; EXEC is ignored by tensor ops; tracked with TENSORcnt.
  {
    const uint64_t boxGA = (uint64_t)(uintptr_t)(box + (size_t)b * 4 * plane +
                                                 (size_t)tileY0 * kW + tileX0);
    const uint32_t boxLDS = lds_addr32(&shBox[0][0][0]);
    if (tid < 32) {
      u32x4 g0;
      g0[0] = 0x1u;                                   // count=1, gather off
      g0[1] = boxLDS;                                 // lds_addr
      g0[2] = (uint32_t)(boxGA & 0xFFFFFFFFu);        // global_addr[31:0]
      g0[3] = ((uint32_t)(boxGA >> 32) & 0x01FFFFFFu) // global_addr[56:32]
              | (0x2u << 30);                         // type=2
      i32x8 g1;
      g1[0] = (int)(2u << 16);             // data_size=2 (4 bytes)
      g1[1] = (int)(192u << 16);           // tensor_dim0[15:0] in [31:16]
      g1[2] = (int)(192u << 16);           // dim0 hi=0 | tensor_dim1[15:0]
      g1[3] = (int)(32u << 16);            // dim1 hi=0 | tile_dim0=32
      g1[4] = (int)(8u | (4u << 16));      // tile_dim1=8 | tile_dim2=4
      g1[5] = (int)192;                    // tensor_dim0_stride lo32
      g1[6] = (int)(36864u << 16);         // stride0 hi=0 | stride1[15:0]
      g1[7] = 0;                           // stride1 hi32 = 0
      i32x4 g2;
      g2[0] = 4;                           // tensor_dim2 = 4 planes
      g2[1] = 0; g2[2] = 0; g2[3] = 0;     // no dim3 / no iterate
      i32x4 g3;
      g3[0] = 0; g3[1] = 0; g3[2] = 0; g3[3] = 0;
      i32x8 g4;                            // extra operand on this toolchain:
      g4[0] = 0; g4[1] = 0; g4[2] = 0;     // zero-filled (reserved/unused)
      g4[3] = 0; g4[4] = 0; g4[5] = 0;
      g4[6] = 0; g4[7] = 0;
      __builtin_amdgcn_tensor_load_to_lds(g0, g1, g2, g3, g4, /*cpol=*/0);
    }
  }

  // ---- per-thread halo descriptors: exactly 2 async loads per channel ----
  const int iA = tid;          // slot 0..255  (< 340 -> always a real halo cell)
  const int iB = tid + 256;    // slot 256..511 (>= 340 -> dummy zero-page load)
  uint64_t baseA, baseB, strA, strB;
  {
    const int ly = iA / kHaloW, lx = iA % kHaloW;
    const int gy = tileY0 + ly - 1, gx = tileX0 + lx - 1;
    if (gy >= 0 && gy < kH && gx >= 0 && gx < kW) {
      baseA = (uint64_t)(uintptr_t)(heat + (size_t)b * kC * plane +
                                    (size_t)gy * kW + gx);
      strA  = (uint64_t)plane * 4u;          // channel stride in bytes
    } else {
      baseA = (uint64_t)(uintptr_t)zeropage; // zero-pad == -inf pad (heat>=0)
      strA  = 0;
    }
  }
  {
    const bool real = iB < kHaloN;
    const int ly = iB / kHaloW, lx = iB % kHaloW;
    const int gy = tileY0 + ly - 1, gx = tileX0 + lx - 1;
    if (real && gy >= 0 && gy < kH && gx >= 0 && gx < kW) {
      baseB = (uint64_t)(uintptr_t)(heat + (size_t)b * kC * plane +
                                    (size_t)gy * kW + gx);
      strB  = (uint64_t)plane * 4u;
    } else {
      baseB = (uint64_t)(uintptr_t)zeropage;
      strB  = 0;
    }
  }

  const uint32_t shbase = lds_addr32(&shHeat[0][0][0]);
  const uint32_t ldsA   = shbase + (uint32_t)iA * 4u;
  const uint32_t ldsB   = shbase + (uint32_t)iB * 4u;

  const int px = tid & (kTX - 1);
  const int py = tid >> 5; // kTX == 32
  const int gx = tileX0 + px;
  const int gy = tileY0 + py;

  uint64_t aA = baseA, aB = baseB;

  // Uniform: every wave issues exactly 2*kCPB = 8 async loads per group.
#define CN_ISSUE_GROUP(BUFOFF)                                             \
  do {                                                                     \
    _Pragma("unroll")                                                      \
    for (int cc = 0; cc < kCPB; ++cc) {                                    \
      const uint64_t gA = aA + (uint64_t)cc * strA;                        \
      const uint64_t gB = aB + (uint64_t)cc * strB;                        \
      const uint32_t lA = ldsA + (BUFOFF) + (uint32_t)cc * kBufBytes;      \
      const uint32_t lB = ldsB + (BUFOFF) + (uint32_t)cc * kBufBytes;      \
      asm volatile("global_load_async_to_lds_b32 %0, %1, off"              \
                   :: "v"(lA), "v"(gA) : "memory");                        \
      asm volatile("global_load_async_to_lds_b32 %0, %1, off"              \
                   :: "v"(lB), "v"(gB) : "memory");                        \
    }                                                                      \
    aA += (uint64_t)kCPB * strA;                                           \
    aB += (uint64_t)kCPB * strB;                                           \
  } while (0)

  CN_ISSUE_GROUP(0u); // channels 0..3 into buffer 0

  const int r0 = py * kHaloW + px; // top-left of 3x3 window in halo tile
  float score = 0.0f;

  for (int g = 0; g < kGroups; ++g) {
    if (g + 1 < kGroups) {
      CN_ISSUE_GROUP((uint32_t)((g + 1) & 1) * kGrpBytes);
      // 8 newer loads outstanding; async loads complete in order, so <=8
      // guarantees group g's 8 loads have landed in LDS.
      asm volatile("s_wait_asynccnt 8" ::: "memory");
    } else {
      asm volatile("s_wait_asynccnt 0" ::: "memory");
    }
    __syncthreads(); // all waves' LDS tiles for group g visible

#pragma unroll
    for (int cc = 0; cc < kCPB; ++cc) {
      const float* t = &shHeat[g & 1][cc][0];
      const float v  = t[r0 + kHaloW + 1]; // center
      float m = fmaxf(fmaxf(t[r0], t[r0 + 1]), t[r0 + 2]);
      m = fmaxf(m, fmaxf(t[r0 + kHaloW], fmaxf(v, t[r0 + kHaloW + 2])));
      m = fmaxf(m, fmaxf(t[r0 + 2 * kHaloW],
                         fmaxf(t[r0 + 2 * kHaloW + 1],
                               t[r0 + 2 * kHaloW + 2])));
      // NMS: keep heat where it equals the 3x3 window max; else 0.
      score = fmaxf(score, (m == v) ? v : 0.0f);
    }

    __syncthreads(); // protect buffer (g&1) before refill at group g+2
  }
#undef CN_ISSUE_GROUP

  // ---- epilogue: box decode (box tile was DMA'd by TDM during the loop) ----
  __builtin_amdgcn_s_wait_tensorcnt(0); // no-op for waves 1..7 (TENSORcnt==0)
  __syncthreads();                      // publish wave 0's completed DMA

  const float dx = shBox[0][py][px];
  const float dy = shBox[1][py][px];
  const float bw = shBox[2][py][px];
  const float bh = shBox[3][py][px];

  const float cx = ((float)gx + dx) * kR;
  const float cy = ((float)gy + dy) * kR;
  const float w2 = bw * kR * 0.5f;
  const float h2 = bh * kR * 0.5f;
  const float limx = (float)kW * kR; // 768
  const float limy = (float)kH * kR; // 768
  const float x1 = fminf(fmaxf(cx - w2, 0.0f), limx);
  const float y1 = fminf(fmaxf(cy - h2, 0.0f), limy);
  const float x2 = fminf(fmaxf(cx + w2, 0.0f), limx);
  const float y2 = fminf(fmaxf(cy + h2, 0.0f), limy);

  const bool keep = score > kThr;
  float4 bxo = keep ? make_float4(x1, y1, x2 - x1, y2 - y1)
                    : make_float4(0.f, 0.f, 0.f, 0.f);
  float2 cto = keep ? make_float2(cx, cy) : make_float2(0.f, 0.f);

  const size_t pix = (size_t)b * plane + (size_t)gy * kW + gx;
  float* outBoxes  = out;
  float* outMask   = out + (size_t)kB * plane * 4;
  float* outScore  = outMask + (size_t)kB * plane;
  float* outCenter = outScore + (size_t)kB * plane;

  reinterpret_cast<float4*>(outBoxes)[pix] = bxo;
  outMask[pix]  = keep ? 1.0f : 0.0f;
  outScore[pix] = keep ? score : 0.0f;
  reinterpret_cast<float2*>(outCenter)[pix] = cto;
}

extern "C" void kernel_launch(void* const* d_in, const int* in_sizes, int n_in,
                              void* d_out, int out_size, void* d_ws, size_t ws_size,
                              hipStream_t stream) {
  (void)in_sizes; (void)n_in; (void)out_size; (void)ws_size;
  const float* heat = (const float*)d_in[0]; // [8,80,192,192]
  const float* box  = (const float*)d_in[1]; // [8,4,192,192]
  float* out = (float*)d_out;
  float* ws  = (float*)d_ws;

  cn_zero_ws<<<dim3(1), dim3(64), 0, stream>>>(ws);

  dim3 grid(kW / kTX, kH / kTY, kB); // 6 x 24 x 8
  dim3 block(256);                   // 8 waves (wave32), 1 pixel/thread
  cn_decode<<<grid, block, 0, stream>>>(heat, box, out, ws);
}